// AdvancedQuantumLayer_3865470567182
// MI455X (gfx1250) — compile-verified
//
#include <hip/hip_runtime.h>
#include <hip/hip_bf16.h>

// CDNA5 (gfx1250) wave32 WMMA types
typedef __attribute__((ext_vector_type(2))) float v2f;
typedef __attribute__((ext_vector_type(8))) float v8f;

#define NQ     16
#define NSTATE 65536      // 2^16 amplitudes
#define DEPTH  3
#define NB     64
#define BLK    256        // 8 wave32 waves

// LDS layout (floats): psi[65536] | M[7*256] | red[256] | th[48] | amp[32]
#define NMAT   7
#define LDS_FLOATS (NSTATE + NMAT * 256 + 256 + 48 + 32)

// D = A(16x4) x B(4x16) + C via v_wmma_f32_16x16x4_f32, K accumulated in 4 chunks.
__device__ __forceinline__ v8f wmma4(v2f a, v2f b, v8f c) {
    return __builtin_amdgcn_wmma_f32_16x16x4_f32(
        /*neg_a=*/false, a, /*neg_b=*/false, b,
        /*c_mod=*/(short)0, c, /*reuse_a=*/false, /*reuse_b=*/false);
}

// Kronecker RY entry: (Ry(th[4g]) (x) ... (x) Ry(th[4g+3]))[j,k]
__device__ __forceinline__ float kron_ry(const float* th16, int g, int j, int k) {
    float p = 1.0f;
#pragma unroll
    for (int t = 0; t < 4; ++t) {
        float h = 0.5f * th16[4 * g + t];
        float c = __cosf(h), s = __sinf(h);
        int jb = (j >> (3 - t)) & 1;
        int kb = (k >> (3 - t)) & 1;
        p *= jb ? (kb ? c : s) : (kb ? -s : c);  // Ry = [[c,-s],[s,c]]
    }
    return p;
}

// Apply a 16x16 matrix along the 4-bit slice at shift S of the state index:
//   out[hi, j, lo] = sum_k M[j,k] psi[hi, k, lo]
// COND:  select M vs M+256 based on the lowest hi bit (column-uniform per tile).
// REMAP: column enumeration with bit0 taken from the tile index (needed for S==0
//        so the condition bit is constant within a 16-column tile).
template <int S, bool COND, bool REMAP>
__device__ __forceinline__ void apply_group(float* __restrict__ psi,
                                            const float* __restrict__ Mbase,
                                            int lane, int wave) {
    const int n  = lane & 15;            // column within tile / A row
    const int kh = (lane >> 4) << 1;     // K half: 0 (lanes 0-15) / 2 (lanes 16-31)
    const int lowmask = (1 << S) - 1;

    v2f a0, a1, a2, a3;
    if (!COND) {  // A is tile-invariant: hoist out of the tile loop
        a0[0] = Mbase[n * 16 + 0  + kh]; a0[1] = Mbase[n * 16 + 1  + kh];
        a1[0] = Mbase[n * 16 + 4  + kh]; a1[1] = Mbase[n * 16 + 5  + kh];
        a2[0] = Mbase[n * 16 + 8  + kh]; a2[1] = Mbase[n * 16 + 9  + kh];
        a3[0] = Mbase[n * 16 + 12 + kh]; a3[1] = Mbase[n * 16 + 13 + kh];
    }

    for (int t = wave; t < 256; t += 8) {  // uniform per wave -> EXEC all ones
        int c    = REMAP ? (((t >> 1) << 5) | (n << 1) | (t & 1)) : (t * 16 + n);
        int base = ((c >> S) << (S + 4)) | (c & lowmask);

        if (COND) {
            const float* A = Mbase + (((c >> S) & 1) << 8);  // tile-uniform select
            a0[0] = A[n * 16 + 0  + kh]; a0[1] = A[n * 16 + 1  + kh];
            a1[0] = A[n * 16 + 4  + kh]; a1[1] = A[n * 16 + 5  + kh];
            a2[0] = A[n * 16 + 8  + kh]; a2[1] = A[n * 16 + 9  + kh];
            a3[0] = A[n * 16 + 12 + kh]; a3[1] = A[n * 16 + 13 + kh];
        }

        v2f b0, b1, b2, b3;  // B 4x16 chunks: element (k, n) at base + (k<<S)
        b0[0] = psi[base + ((0  + kh) << S)]; b0[1] = psi[base + ((1  + kh) << S)];
        b1[0] = psi[base + ((4  + kh) << S)]; b1[1] = psi[base + ((5  + kh) << S)];
        b2[0] = psi[base + ((8  + kh) << S)]; b2[1] = psi[base + ((9  + kh) << S)];
        b3[0] = psi[base + ((12 + kh) << S)]; b3[1] = psi[base + ((13 + kh) << S)];

        v8f acc = {};
        acc = wmma4(a0, b0, acc);
        acc = wmma4(a1, b1, acc);
        acc = wmma4(a2, b2, acc);
        acc = wmma4(a3, b3, acc);

        // D: lanes0-15 VGPR r -> row r; lanes16-31 -> row r+8. Tile-local, no race.
        const int rb = (lane >> 4) * 8;
#pragma unroll
        for (int r = 0; r < 8; ++r)
            psi[base + ((rb + r) << S)] = acc[r];
    }
}

// Final fused pass: apply the deferred R0 (group 0, S=12) and reduce.
// Group 0's row bit 3 == state bit 15 == qubit 0, so rows 8..15 (lanes 16-31)
// are exactly the qubit0==1 amplitudes: return sum of their squares (read-only,
// no store of D).
__device__ __forceinline__ float apply_r0_reduce(const float* __restrict__ psi,
                                                 const float* __restrict__ Mbase,
                                                 int lane, int wave) {
    const int S  = 12;
    const int n  = lane & 15;
    const int kh = (lane >> 4) << 1;
    const float msk = (lane >= 16) ? 1.0f : 0.0f;  // rows 8..15 only

    v2f a0, a1, a2, a3;
    a0[0] = Mbase[n * 16 + 0  + kh]; a0[1] = Mbase[n * 16 + 1  + kh];
    a1[0] = Mbase[n * 16 + 4  + kh]; a1[1] = Mbase[n * 16 + 5  + kh];
    a2[0] = Mbase[n * 16 + 8  + kh]; a2[1] = Mbase[n * 16 + 9  + kh];
    a3[0] = Mbase[n * 16 + 12 + kh]; a3[1] = Mbase[n * 16 + 13 + kh];

    float local = 0.0f;
    for (int t = wave; t < 256; t += 8) {
        int c    = t * 16 + n;
        int base = ((c >> S) << (S + 4)) | (c & ((1 << S) - 1));

        v2f b0, b1, b2, b3;
        b0[0] = psi[base + ((0  + kh) << S)]; b0[1] = psi[base + ((1  + kh) << S)];
        b1[0] = psi[base + ((4  + kh) << S)]; b1[1] = psi[base + ((5  + kh) << S)];
        b2[0] = psi[base + ((8  + kh) << S)]; b2[1] = psi[base + ((9  + kh) << S)];
        b3[0] = psi[base + ((12 + kh) << S)]; b3[1] = psi[base + ((13 + kh) << S)];

        v8f acc = {};
        acc = wmma4(a0, b0, acc);
        acc = wmma4(a1, b1, acc);
        acc = wmma4(a2, b2, acc);
        acc = wmma4(a3, b3, acc);

        float sq = 0.0f;
#pragma unroll
        for (int r = 0; r < 8; ++r) sq += acc[r] * acc[r];
        local += msk * sq;
    }
    return local;
}

// 16-qubit state-vector simulator, one workgroup per batch element; state lives
// in LDS (256KB of the 320KB WGP). Per depth = 6 WMMA passes:
//   G0 = P (d=0) or P*R0_prev (deferred R0) | cond[P,PX](c3+A1) |
//   cond[P,PX](c7+A2) | cond[R3P,R3PX](c11+A3+R3) | R1 | R2
// Final: read-only R0+|.|^2 reduction straight from WMMA accumulators.
__global__ __launch_bounds__(BLK) void qsim16_wmma_kernel(
    const float* __restrict__ x,       // [64,16]
    const float* __restrict__ thetas,  // [3,16]
    float* __restrict__ out)           // [64,1]
{
    extern __shared__ float lds[];
    float* psi = lds;                    // 65536
    float* M   = lds + NSTATE;           // 7 * 256
    float* red = M + NMAT * 256;         // 256
    float* th  = red + 256;              // 48
    float* amp = th + 48;                // 32: amp[2q+bit]

    const int b    = blockIdx.x;
    const int tid  = threadIdx.x;
    const int lane = tid & 31;
    const int wave = tid >> 5;

    // ---- load params ----
    if (tid < DEPTH * NQ) th[tid] = thetas[tid];
    if (tid < NQ) {
        float h = 0.5f * x[b * NQ + tid];
        amp[2 * tid + 0] = __cosf(h);
        amp[2 * tid + 1] = __sinf(h);
    }

    // ---- depth-invariant permutation matrices ----
    // Intra-group CX chain: basis map f(m) = prefix-XOR from group MSB; f^-1 = gray.
    const int j   = tid >> 4;
    const int k   = tid & 15;
    const int fk  = (k ^ (k >> 1) ^ (k >> 2) ^ (k >> 3)) & 15;
    const int kx  = k ^ 8;                                   // cross-CX flips top bit
    const int fkx = (kx ^ (kx >> 1) ^ (kx >> 2) ^ (kx >> 3)) & 15;
    const int gj  = (j ^ (j >> 1)) & 15;                     // f^-1(j)
    M[0 * 256 + tid] = (j == fk)  ? 1.0f : 0.0f;   // P
    M[1 * 256 + tid] = (j == fkx) ? 1.0f : 0.0f;   // P * X8
    __syncthreads();

    // ---- init product state: psi[i] = prod_q amp[q][bit_q(i)], qubit0 = MSB ----
    for (int i = tid; i < NSTATE; i += BLK) {
        float p = 1.0f;
#pragma unroll
        for (int q = 0; q < NQ; ++q)
            p *= amp[2 * q + ((i >> (15 - q)) & 1)];
        psi[i] = p;
    }

    for (int d = 0; d < DEPTH; ++d) {
        // previous depth's passes must be done reading M before we rebuild it
        __syncthreads();
        const float* thd = th + d * NQ;
        M[2 * 256 + tid] = kron_ry(thd, 3, j, fk);   // R3 * P
        M[3 * 256 + tid] = kron_ry(thd, 3, j, fkx);  // R3 * P * X8
        M[4 * 256 + tid] = kron_ry(thd, 1, j, k);    // R1
        M[5 * 256 + tid] = kron_ry(thd, 2, j, k);    // R2
        if (d > 0)                                    // G0 = P * R0_prev
            M[6 * 256 + tid] = kron_ry(th + (d - 1) * NQ, 0, gj, k);
        const float* G0m = (d == 0) ? (M + 0 * 256) : (M + 6 * 256);

        __syncthreads(); apply_group<12, false, false>(psi, G0m,         lane, wave); // A0 (*R0_prev)
        __syncthreads(); apply_group< 8, true , false>(psi, M + 0 * 256, lane, wave); // c3 + A1
        __syncthreads(); apply_group< 4, true , false>(psi, M + 0 * 256, lane, wave); // c7 + A2
        __syncthreads(); apply_group< 0, true , true >(psi, M + 2 * 256, lane, wave); // c11 + A3 + R3
        __syncthreads(); apply_group< 8, false, false>(psi, M + 4 * 256, lane, wave); // R1
        __syncthreads(); apply_group< 4, false, false>(psi, M + 5 * 256, lane, wave); // R2
    }

    // ---- deferred R0 of last depth fused with p1 reduction (read-only) ----
    __syncthreads();
    M[2 * 256 + tid] = kron_ry(th + (DEPTH - 1) * NQ, 0, j, k);  // R0_last
    __syncthreads();
    red[tid] = apply_r0_reduce(psi, M + 2 * 256, lane, wave);
    __syncthreads();
    for (int off = BLK / 2; off > 0; off >>= 1) {
        if (tid < off) red[tid] += red[tid + off];
        __syncthreads();
    }
    if (tid == 0) out[b] = red[0];
}

extern "C" void kernel_launch(void* const* d_in, const int* in_sizes, int n_in,
                              void* d_out, int out_size, void* d_ws, size_t ws_size,
                              hipStream_t stream) {
    const float* x      = (const float*)d_in[0];  // [64,16] f32
    const float* thetas = (const float*)d_in[1];  // [3,16]  f32
    float* out          = (float*)d_out;          // [64,1]  f32

    const size_t shbytes = LDS_FLOATS * sizeof(float); // ~271 KB of 320 KB WGP LDS
    hipFuncSetAttribute((const void*)qsim16_wmma_kernel,
                        hipFuncAttributeMaxDynamicSharedMemorySize, (int)shbytes);
    qsim16_wmma_kernel<<<NB, BLK, shbytes, stream>>>(x, thetas, out);
}